// NeuralHawkes_17051020165452
// MI455X (gfx1250) — compile-verified
//
#include <hip/hip_runtime.h>
#include <hip/hip_bf16.h>
#include <math.h>

// Neural Hawkes CT-LSTM, fully fused for MI455X (gfx1250, wave32, WMMA).
// B=256, G=1, T2=1026 (Tp1=1025), S=1024, H=32, EVENT_NUM=32.

#define B_    256
#define T2_   1026
#define TP1   1025
#define S_    1024
#define H_    32
#define KDIM  64     // 2*H
#define NPRE  224    // 7*H
#define EVN   32
#define EPS_  2.220446049250313e-16f

typedef float v2f __attribute__((ext_vector_type(2)));
typedef float v8f __attribute__((ext_vector_type(8)));

__device__ __forceinline__ float sigmoidf_(float x) { return 1.0f / (1.0f + __expf(-x)); }
__device__ __forceinline__ float softplusf_(float x) {
    // numerically stable softplus, matches jax.nn.softplus
    return fmaxf(x, 0.0f) + log1pf(__expf(-fabsf(x)));
}

__global__ void nh_init_kernel(float* __restrict__ lamAcc) {
    lamAcc[threadIdx.x] = 0.0f;   // 256 per-batch lambda-integral accumulators
}

// One workgroup owns 16 batch rows and runs the whole 1025-step scan with the
// epilogue fused in. 512 threads = 16 waves; waves 0..13 each own one 16x16
// N-tile of the 16x224 pre-activation GEMM via V_WMMA_F32_16X16X4_F32.
__global__ __launch_bounds__(512)
void nh_scan_kernel(const int*   __restrict__ event,
                    const float* __restrict__ dtime,
                    const float* __restrict__ dts,      // dtime_sampling (B,S) flat
                    const float* __restrict__ msamp,    // mask_sampling  (B,S) flat
                    const float* __restrict__ Emb,      // (VOCAB, H)
                    const float* __restrict__ W,        // (64, 224)
                    const float* __restrict__ bias,     // (224)
                    const float* __restrict__ Wl,       // (32, 32)
                    float* __restrict__ lamOut,         // d_out + 2, (256*1024*32)
                    float* __restrict__ lamAcc,         // ws[0:256]
                    float* __restrict__ loglikOut,      // ws[256:512]
                    float* __restrict__ nevOut)         // ws[512:768]
{
    __shared__ float sW  [KDIM * NPRE];   // 57344 B
    __shared__ float sWl [EVN  * H_  ];   //  4096 B
    __shared__ float sB  [NPRE];          //   896 B
    __shared__ float sX  [16 * KDIM];     //  4096 B  [emb | h]
    __shared__ float sPre[16 * NPRE];     // 14336 B
    __shared__ float sC  [16 * H_];       //  2048 B
    __shared__ float sCb [16 * H_];       //  2048 B
    __shared__ float sSh [16 * H_];       //  2048 B
    __shared__ float sLam[16 * EVN];      //  2048 B   (~89 KB total; WGP has 320 KB)

    const int tid = threadIdx.x;
    const int b0  = blockIdx.x * 16;

    // load constants into LDS
    for (int i = tid; i < KDIM * NPRE; i += 512) sW[i]  = W[i];
    for (int i = tid; i < EVN * H_;    i += 512) sWl[i] = Wl[i];
    for (int i = tid; i < NPRE;        i += 512) sB[i]  = bias[i];

    // init carry state (c = cbar = h = 0)
    {
        int m = tid >> 5, j = tid & 31;
        sC [m * 32 + j] = 0.0f;
        sCb[m * 32 + j] = 0.0f;
        sX [m * 64 + 32 + j] = 0.0f;
    }
    __syncthreads();

    const int wave = tid >> 5;
    const int lane = tid & 31;
    const int ln15 = lane & 15;
    const int half = lane >> 4;

    float ll_acc = 0.0f;   // per-row log-likelihood (valid for tid < 16)
    float ne_acc = 0.0f;   // per-row event count

    for (int t = 0; t < TP1; ++t) {
        // ---- P0: embedding lookup for this step -------------------------
        {
            int m = tid >> 5, j = tid & 31;
            int ev = event[(b0 + m) * T2_ + t];
            sX[m * 64 + j] = Emb[ev * H_ + j];
        }
        __syncthreads();

        // ---- P1: pre = [x,h] @ W + bias via f32 WMMA --------------------
        if (wave < 14) {                       // wave-uniform branch: EXEC stays all-1
            const int n0 = wave * 16;
            v8f acc;
            const float bv = sB[n0 + ln15];
            #pragma unroll
            for (int v = 0; v < 8; ++v) acc[v] = bv;
            #pragma unroll
            for (int kb = 0; kb < 16; ++kb) {  // K = 64 in steps of 4
                const int ka = kb * 4 + half * 2;
                v2f a, bf;
                a[0]  = sX[ln15 * 64 + ka + 0];
                a[1]  = sX[ln15 * 64 + ka + 1];
                bf[0] = sW[(ka + 0) * NPRE + n0 + ln15];
                bf[1] = sW[(ka + 1) * NPRE + n0 + ln15];
                acc = __builtin_amdgcn_wmma_f32_16x16x4_f32(
                          false, a, false, bf, (short)0, acc, false, false);
            }
            #pragma unroll
            for (int v = 0; v < 8; ++v)
                sPre[(v + 8 * half) * NPRE + n0 + ln15] = acc[v];
        }
        __syncthreads();

        // ---- P2: gates, state update, fused sampled-hidden --------------
        {
            const int m = tid >> 5, j = tid & 31;
            const float* p = &sPre[m * NPRE];
            float gi  = sigmoidf_(p[j]);
            float gf  = sigmoidf_(p[j + 32]);
            float gz  = tanhf    (p[j + 64]);
            float go  = sigmoidf_(p[j + 96]);
            float gib = sigmoidf_(p[j + 128]);
            float gfb = sigmoidf_(p[j + 160]);
            float del = softplusf_(p[j + 192]);
            float cm  = sC [m * 32 + j];
            float cbm = sCb[m * 32 + j];
            float c   = gf  * cm  + gi  * gz;      // pre-decay cell (output "c")
            float cb  = gfb * cbm + gib * gz;      // cbar
            float dt  = dtime[(b0 + m) * T2_ + t + 1];
            float cn  = cb + (c - cb) * __expf(-del * dt);
            float hn  = go * tanhf(cn);
            sC [m * 32 + j] = cn;
            sCb[m * 32 + j] = cb;
            sX [m * 64 + 32 + j] = hn;             // h for next step + loglik dot

            // flat-map: row r = b*1025 + t -> output slot (r/1024, r%1024)
            int   r  = (b0 + m) * TP1 + t;
            float sh = 0.0f;
            if (r < B_ * S_) {
                float dv   = dts[r];
                float cdec = cb + (c - cb) * __expf(-del * dv);
                sh = go * tanhf(cdec);
            }
            sSh[m * 32 + j] = sh;
        }
        __syncthreads();

        // ---- P3: lambda_sample = softplus(sh @ Wl^T), write output ------
        {
            const int m = tid >> 5, e = tid & 31;
            const int r = (b0 + m) * TP1 + t;
            const float* sh = &sSh[m * 32];
            const float* wl = &sWl[e * 32];
            float dot = 0.0f;
            #pragma unroll
            for (int j = 0; j < 32; ++j) dot += sh[j] * wl[j];
            float lam = softplusf_(dot);
            sLam[m * 32 + e] = lam;
            if (r < B_ * S_) lamOut[r * 32 + e] = lam;
        }
        __syncthreads();

        // ---- P4: per-row reductions (loglik + lambda integral) ----------
        if (tid < 16) {
            const int m = tid;
            const int r = (b0 + m) * TP1 + t;
            if (r < B_ * S_) {
                float rs = 0.0f;
                #pragma unroll
                for (int e = 0; e < 32; ++e) rs += sLam[m * 32 + e];
                atomicAdd(&lamAcc[r >> 10], rs * msamp[r]);
            }
            int tgt = event[(b0 + m) * T2_ + t + 1];
            if (tgt < EVN) {
                const float* h  = &sX[m * 64 + 32];
                const float* wl = &sWl[tgt * 32];
                float dot = 0.0f;
                #pragma unroll
                for (int j = 0; j < 32; ++j) dot += h[j] * wl[j];
                ll_acc += __logf(softplusf_(dot) + EPS_);
                ne_acc += 1.0f;
            }
        }
        __syncthreads();
    }

    if (tid < 16) {
        loglikOut[b0 + tid] = ll_acc;
        nevOut  [b0 + tid] = ne_acc;
    }
}

__global__ void nh_finalize_kernel(const float* __restrict__ msamp,
                                   const float* __restrict__ duration, // (B,G)
                                   const float* __restrict__ lamAcc,
                                   const float* __restrict__ loglik,
                                   const float* __restrict__ nev,
                                   float* __restrict__ out)            // d_out
{
    __shared__ float sObj[256];
    __shared__ float sNe [256];
    const int b = threadIdx.x;
    float denom = 0.0f;
    for (int s = 0; s < S_; ++s) denom += msamp[b * S_ + s];
    float integral = (lamAcc[b] / denom) * duration[b];
    sObj[b] = loglik[b] - integral;   // logP per batch
    sNe [b] = nev[b];
    __syncthreads();
    for (int off = 128; off > 0; off >>= 1) {
        if (b < off) { sObj[b] += sObj[b + off]; sNe[b] += sNe[b + off]; }
        __syncthreads();
    }
    if (b == 0) { out[0] = -sObj[0]; out[1] = sNe[0]; }
}

extern "C" void kernel_launch(void* const* d_in, const int* in_sizes, int n_in,
                              void* d_out, int out_size, void* d_ws, size_t ws_size,
                              hipStream_t stream) {
    const int*   event    = (const int*)  d_in[0];
    const float* dtime    = (const float*)d_in[1];
    const float* duration = (const float*)d_in[2];
    const float* dts      = (const float*)d_in[3];
    // d_in[4] index_of_hidden_sampling: unused by the reference computation
    const float* msamp    = (const float*)d_in[5];
    const float* Emb      = (const float*)d_in[6];
    const float* W        = (const float*)d_in[7];
    const float* bias     = (const float*)d_in[8];
    const float* Wl       = (const float*)d_in[9];

    float* out    = (float*)d_out;
    float* ws     = (float*)d_ws;
    float* lamAcc = ws;         // 256 f32
    float* loglik = ws + 256;   // 256 f32
    float* nev    = ws + 512;   // 256 f32

    nh_init_kernel<<<1, 256, 0, stream>>>(lamAcc);
    nh_scan_kernel<<<16, 512, 0, stream>>>(event, dtime, dts, msamp, Emb, W, bias, Wl,
                                           out + 2, lamAcc, loglik, nev);
    nh_finalize_kernel<<<1, 256, 0, stream>>>(msamp, duration, lamAcc, loglik, nev, out);
}